// TrafficLightEncoder_52355651338940
// MI455X (gfx1250) — compile-verified
//
#include <hip/hip_runtime.h>
#include <hip/hip_bf16.h>

// ---------------------------------------------------------------------------
// Problem constants (from the reference)
// ---------------------------------------------------------------------------
#define N_SC   4
#define N_TL   256
#define N_MP   4096
#define HIDDEN 512
#define KNN    36
#define NFREQ  32
#define FEAT_REAL 130          // 4*NFREQ + 2
#define FEAT_PAD  132          // padded to multiple of 4 for K-loop
#define KPAIRS    (FEAT_PAD/2) // 66 row-pairs in the padded weight
#define BIGD   1.0e6f
#define ROWS_SEG (N_SC * N_TL * KNN)   // 36864 rows per segment (tt / tm)
#define GEMM_ROWS_PER_BLK 64

typedef float v2f __attribute__((ext_vector_type(2)));
typedef float v8f __attribute__((ext_vector_type(8)));

__device__ __forceinline__ float wrap_angle(float a) {
    const float PI  = 3.14159265358979323846f;
    const float TPI = 6.28318530717958647692f;
    float w = fmodf(a + PI, TPI);
    if (w < 0.f) w += TPI;          // python-style mod (always non-negative)
    return w - PI;
}

// ---------------------------------------------------------------------------
// Kernel 0: zero-pad + pair-interleave W.
//   Wpair[p][col][2] = { W[2p][col], W[2p+1][col] }, rows >= 130 are zero.
// Makes every B fragment in the GEMM a single coalesced global_load_b64
// and removes all K-edge guards (no EXEC churn around WMMA).
// ---------------------------------------------------------------------------
__global__ void __launch_bounds__(256)
pad_w(const float* __restrict__ W, float* __restrict__ Wpair) {
    const int idx = blockIdx.x * 256 + threadIdx.x;     // FEAT_PAD*HIDDEN total
    if (idx >= FEAT_PAD * HIDDEN) return;
    const int p   = idx >> 10;          // / (HIDDEN*2)
    const int r   = idx & 1023;
    const int col = r >> 1;
    const int j   = r & 1;
    const int row = 2 * p + j;
    Wpair[idx] = (row < FEAT_REAL) ? W[(size_t)row * HIDDEN + col] : 0.f;
}

// ---------------------------------------------------------------------------
// Kernel 1: tl->tl top-36 (256 candidates). One block per (s,tl).
// u64 keys = (f32 dist bits << 32) | idx  -> min gives smallest dist with
// smallest-index tie-break, exactly matching lax.top_k on -dist.
// ---------------------------------------------------------------------------
__global__ void __launch_bounds__(256)
topk_tt(const float* __restrict__ tl_pose,
        const unsigned char* __restrict__ tl_valid,
        float* __restrict__ rpe_tt /* [S*N_TL*36*3] */) {
    __shared__ unsigned long long keys[N_TL];
    __shared__ unsigned long long red[256];
    __shared__ int winners[KNN];

    const int b = blockIdx.x;            // s*N_TL + t
    const int s = b >> 8;
    const int i = threadIdx.x;

    const float sx = tl_pose[b*3+0], sy = tl_pose[b*3+1], syaw = tl_pose[b*3+2];
    const bool sinv = (tl_valid[b] == 0);

    {
        const float* tp = tl_pose + ((size_t)s*N_TL + i)*3;
        float dx = tp[0]-sx, dy = tp[1]-sy;
        float d  = sqrtf(dx*dx + dy*dy);
        bool inv = sinv || (tl_valid[s*N_TL + i] == 0);
        if (inv) d = BIGD;
        keys[i] = ((unsigned long long)__float_as_uint(d) << 32) | (unsigned)i;
    }
    __syncthreads();

    for (int it = 0; it < KNN; ++it) {
        red[i] = keys[i];
        __syncthreads();
        for (int off = 128; off > 0; off >>= 1) {
            if (i < off) {
                unsigned long long o = red[i+off];
                if (o < red[i]) red[i] = o;
            }
            __syncthreads();
        }
        if (i == 0) {
            int cand = (int)(red[0] & 0xffffffffull);
            winners[it] = cand;
            keys[cand] = ~0ull;          // remove winner
        }
        __syncthreads();
    }

    if (i < KNN) {
        const int cand = winners[i];
        const float* tp = tl_pose + ((size_t)s*N_TL + cand)*3;
        float dx = tp[0]-sx, dy = tp[1]-sy;
        float cy = cosf(syaw), sn = sinf(syaw);
        float lx =  cy*dx + sn*dy;
        float ly = -sn*dx + cy*dy;
        float dyaw = wrap_angle(tp[2] - syaw);
        float* o = rpe_tt + ((size_t)b*KNN + i)*3;
        o[0]=lx; o[1]=ly; o[2]=dyaw;
    }
}

// ---------------------------------------------------------------------------
// Kernel 2: tl->mp top-36 (4096 candidates). One block per (s,tl).
// ---------------------------------------------------------------------------
__global__ void __launch_bounds__(256)
topk_tm(const float* __restrict__ tl_pose,
        const unsigned char* __restrict__ tl_valid,
        const float* __restrict__ mp_pose,
        const unsigned char* __restrict__ mp_invalid,
        int*   __restrict__ idx_tm /* [S*N_TL*36] */,
        float* __restrict__ rpe_tm /* [S*N_TL*36*3] */) {
    __shared__ unsigned long long keys[N_MP];     // 32 KB
    __shared__ unsigned long long red[256];
    __shared__ int winners[KNN];

    const int b = blockIdx.x;
    const int s = b >> 8;
    const int i = threadIdx.x;

    const float sx = tl_pose[b*3+0], sy = tl_pose[b*3+1], syaw = tl_pose[b*3+2];
    const bool sinv = (tl_valid[b] == 0);

    for (int c = i; c < N_MP; c += 256) {
        const float* tp = mp_pose + ((size_t)s*N_MP + c)*3;
        float dx = tp[0]-sx, dy = tp[1]-sy;
        float d  = sqrtf(dx*dx + dy*dy);
        bool inv = sinv || (mp_invalid[s*N_MP + c] != 0);
        if (inv) d = BIGD;
        keys[c] = ((unsigned long long)__float_as_uint(d) << 32) | (unsigned)c;
    }
    __syncthreads();

    for (int it = 0; it < KNN; ++it) {
        unsigned long long loc = ~0ull;
        for (int c = i; c < N_MP; c += 256) {
            unsigned long long k = keys[c];
            if (k < loc) loc = k;
        }
        red[i] = loc;
        __syncthreads();
        for (int off = 128; off > 0; off >>= 1) {
            if (i < off) {
                unsigned long long o = red[i+off];
                if (o < red[i]) red[i] = o;
            }
            __syncthreads();
        }
        if (i == 0) {
            int cand = (int)(red[0] & 0xffffffffull);
            winners[it] = cand;
            keys[cand] = ~0ull;
        }
        __syncthreads();
    }

    if (i < KNN) {
        const int cand = winners[i];
        const float* tp = mp_pose + ((size_t)s*N_MP + cand)*3;
        float dx = tp[0]-sx, dy = tp[1]-sy;
        float cy = cosf(syaw), sn = sinf(syaw);
        float lx =  cy*dx + sn*dy;
        float ly = -sn*dx + cy*dy;
        float dyaw = wrap_angle(tp[2] - syaw);
        idx_tm[(size_t)b*KNN + i] = cand;
        float* o = rpe_tm + ((size_t)b*KNN + i)*3;
        o[0]=lx; o[1]=ly; o[2]=dyaw;
    }
}

// ---------------------------------------------------------------------------
// Kernel 3: feature-row gathers (out col 0 = tl_token_attr, cols 37..72 = knn
// map features). One block per (s,tl,row); float4 (B128) streaming copies.
// ---------------------------------------------------------------------------
__global__ void __launch_bounds__(128)
gather_rows(const float* __restrict__ mp_feat,
            const int*   __restrict__ tl_attr,
            const int*   __restrict__ idx_tm,
            float* __restrict__ out) {
    const int b  = blockIdx.x;
    const int st = b / 37;               // s*N_TL + t
    const int k  = b - st * 37;
    const int s  = st >> 8;

    int src, dcol;
    if (k == 0) { src = tl_attr[st];                  dcol = 0; }
    else        { src = idx_tm[(size_t)st*KNN + k-1]; dcol = 36 + k; }

    const float4* sp = (const float4*)(mp_feat + ((size_t)s*N_MP + src)*HIDDEN);
    float4*       dp = (float4*)(out + ((size_t)st*109 + dcol)*HIDDEN);
    dp[threadIdx.x] = sp[threadIdx.x];   // 128 * float4 = 512 floats
}

// ---------------------------------------------------------------------------
// Kernel 4: sinusoidal feature build + f32 WMMA GEMM projection.
//   segment 0: rpe_tt -> out cols 1..36 ; segment 1: rpe_tm -> cols 73..108
// Block = 64 rows x 512 cols, 256 threads (8 waves). Features [64][132] in
// LDS; inner loop = ds_load(A pair) + global_load_b64(B pair) + wmma,
// branch-free, EXEC all-ones throughout.
// ---------------------------------------------------------------------------
__global__ void __launch_bounds__(256)
rpe_gemm(const float* __restrict__ rpe_all,   // [2 * ROWS_SEG * 3]
         const float* __restrict__ Wpair,     // [66][512][2], zero-padded
         const float* __restrict__ bias,      // [512]
         float* __restrict__ out) {
    __shared__ float poseS[GEMM_ROWS_PER_BLK * 3];
    __shared__ float featS[GEMM_ROWS_PER_BLK * FEAT_PAD];   // ~33.8 KB
    __shared__ float biasS[HIDDEN];
    __shared__ int   obase[GEMM_ROWS_PER_BLK];

    const int tid = threadIdx.x;
    const int seg = blockIdx.y;                              // 0 = tt, 1 = tm
    const int rowBase = blockIdx.x * GEMM_ROWS_PER_BLK;      // row within segment

    // stage poses + bias
    const float* rpe = rpe_all + ((size_t)seg * ROWS_SEG + rowBase) * 3;
    for (int i = tid; i < GEMM_ROWS_PER_BLK * 3; i += 256) poseS[i] = rpe[i];
    biasS[tid]       = bias[tid];
    biasS[tid + 256] = bias[tid + 256];

    // per-row output base offsets
    if (tid < GEMM_ROWS_PER_BLK) {
        int g  = rowBase + tid;
        int st = g / KNN;
        int j  = g - st * KNN;
        int colbase = seg ? 73 : 1;
        obase[tid] = (st * 109 + colbase + j) * HIDDEN;
    }
    __syncthreads();

    // features: [sin(x*2^i), sin(y*2^i), cos(x*2^i), cos(y*2^i), cos(yaw), sin(yaw), 0, 0]
    for (int e = tid; e < GEMM_ROWS_PER_BLK * FEAT_PAD; e += 256) {
        int row = e / FEAT_PAD;
        int col = e - row * FEAT_PAD;
        float lx  = poseS[row*3+0];
        float ly  = poseS[row*3+1];
        float yaw = poseS[row*3+2];
        float val;
        if (col < 128) {
            int band = col & 31;
            float freq  = __uint_as_float((unsigned)(127 + band) << 23);  // 2^band
            float coord = (col & 32) ? ly : lx;
            float t = coord * freq;
            val = (col < 64) ? sinf(t) : cosf(t);
        } else if (col == 128) val = cosf(yaw);
        else if (col == 129)   val = sinf(yaw);
        else                   val = 0.f;
        featS[e] = val;
    }
    __syncthreads();

    const int wave  = tid >> 5;
    const int lane  = tid & 31;
    const int hs    = lane >> 4;      // half-select (K pair / M+8)
    const int l     = lane & 15;
    const int mtile = wave >> 1;      // 4 M-tiles of 16 rows
    const int nhalf = wave & 1;       // 2 waves split the 32 N-tiles

    for (int nt = 0; nt < 16; ++nt) {
        const int ntile = nhalf * 16 + nt;
        const int colg  = ntile * 16 + l;
        const float bv  = biasS[colg];
        v8f c = { bv, bv, bv, bv, bv, bv, bv, bv };

        // per-lane base pointers for this tile
        const float* wp = Wpair + (size_t)hs * (HIDDEN * 2) + (size_t)colg * 2;
        const int    ab = (mtile * 16 + l) * FEAT_PAD + hs * 2;

        for (int k0 = 0; k0 < FEAT_PAD; k0 += 4) {
            // A fragment (16x4 f32): lanes 0-15 K=k0..k0+1, lanes 16-31 K=k0+2..k0+3
            v2f a;
            a[0] = featS[ab + k0];
            a[1] = featS[ab + k0 + 1];
            // B fragment: one b64 load of the (row k, row k+1) pair
            const float2 bb = *(const float2*)(wp + (size_t)(k0 >> 1) * (HIDDEN * 2));
            v2f bfr; bfr[0] = bb.x; bfr[1] = bb.y;
            c = __builtin_amdgcn_wmma_f32_16x16x4_f32(
                    false, a, false, bfr, (short)0, c, false, false);
        }

        // D layout: VGPR r -> row r (lanes 0-15) / row r+8 (lanes 16-31), col = l
        #pragma unroll
        for (int r = 0; r < 8; ++r) {
            int rowL = mtile * 16 + hs * 8 + r;
            out[(size_t)obase[rowL] + colg] = c[r];
        }
    }
}

// ---------------------------------------------------------------------------
// Launch
// ---------------------------------------------------------------------------
extern "C" void kernel_launch(void* const* d_in, const int* in_sizes, int n_in,
                              void* d_out, int out_size, void* d_ws, size_t ws_size,
                              hipStream_t stream) {
    const unsigned char* tl_valid  = (const unsigned char*)d_in[0];
    const int*           tl_attr   = (const int*)d_in[1];
    const float*         tl_pose   = (const float*)d_in[2];
    const unsigned char* mp_inv    = (const unsigned char*)d_in[3];
    const float*         mp_feat   = (const float*)d_in[4];
    const float*         mp_pose   = (const float*)d_in[5];
    const float*         W_rpe     = (const float*)d_in[6];
    const float*         b_rpe     = (const float*)d_in[7];
    float*               out       = (float*)d_out;

    // workspace: rpe_tt | rpe_tm | idx_tm | Wpair
    float* rpe_all = (float*)d_ws;
    float* rpe_tt  = rpe_all;
    float* rpe_tm  = rpe_all + (size_t)ROWS_SEG * 3;
    int*   idx_tm  = (int*)(rpe_all + (size_t)2 * ROWS_SEG * 3);
    float* Wpair   = (float*)(idx_tm + ROWS_SEG);

    pad_w<<<(FEAT_PAD * HIDDEN + 255) / 256, 256, 0, stream>>>(W_rpe, Wpair);
    topk_tt<<<N_SC * N_TL, 256, 0, stream>>>(tl_pose, tl_valid, rpe_tt);
    topk_tm<<<N_SC * N_TL, 256, 0, stream>>>(tl_pose, tl_valid, mp_pose, mp_inv,
                                             idx_tm, rpe_tm);
    gather_rows<<<N_SC * N_TL * 37, 128, 0, stream>>>(mp_feat, tl_attr, idx_tm, out);

    dim3 ggrid(ROWS_SEG / GEMM_ROWS_PER_BLK, 2);   // (576, 2)
    rpe_gemm<<<ggrid, 256, 0, stream>>>(rpe_all, Wpair, b_rpe, out);
}